// VectorQuantizer_4097398800431
// MI455X (gfx1250) — compile-verified
//
#include <hip/hip_runtime.h>

typedef float v2f __attribute__((ext_vector_type(2)));
typedef float v8f __attribute__((ext_vector_type(8)));

#define NUM_CODES 1024
#define CODE_DIM  64
#define HW        4096              // H*W
#define X_ELEMS   8388608           // 32*64*64*64
#define LOSS_OFF  X_ELEMS
#define IDX_OFF   (X_ELEMS + 1)

// ---------------------------------------------------------------------------
// Kernel 0: precompute |c|^2 for all codes into workspace; zero the loss slot
// (must be re-zeroed every launch for determinism).
// ---------------------------------------------------------------------------
__global__ void vq_c2_kernel(const float* __restrict__ cb,
                             float* __restrict__ c2,
                             float* __restrict__ out) {
    int j = blockIdx.x * blockDim.x + threadIdx.x;
    if (j == 0) out[LOSS_OFF] = 0.0f;
    if (j < NUM_CODES) {
        const float4* row = (const float4*)(cb + j * CODE_DIM);
        float s = 0.0f;
#pragma unroll
        for (int i = 0; i < CODE_DIM / 4; ++i) {
            float4 v = row[i];
            s += v.x * v.x + v.y * v.y + v.z * v.z + v.w * v.w;
        }
        c2[j] = s;
    }
}

// ---- B-tile (16 codes x 64 dims) + |c|^2 loader for one 16-code chunk ----
__device__ __forceinline__ void load_chunk(const float* __restrict__ cb,
                                           const float* __restrict__ c2,
                                           int c, int m, int h,
                                           v2f (&bm)[16], float& cn2) {
    const int code = (c << 4) + m;
    const float* cbr = cb + code * CODE_DIM + 2 * h;   // rows contiguous -> b64
#pragma unroll
    for (int kk = 0; kk < 16; ++kk)
        bm[kk] = *(const v2f*)(cbr + 4 * kk);
    cn2 = c2[code];
}

// ---- 16-WMMA chain + branchless running argmin for one chunk ----
__device__ __forceinline__ void process_chunk(int c, int m,
                                              const v2f (&a)[16],
                                              const v2f (&bm)[16], float cn2,
                                              float (&bestD)[8], int (&bestI)[8]) {
    const int code = (c << 4) + m;
    v8f acc;
#pragma unroll
    for (int r = 0; r < 8; ++r) acc[r] = cn2;           // seed C with |c|^2
#pragma unroll
    for (int kk = 0; kk < 16; ++kk) {
        // acc = (-2Z) x C^T + acc ; 8 args: negA,A,negB,B,cmod,C,reuseA,reuseB
        acc = __builtin_amdgcn_wmma_f32_16x16x4_f32(
            false, a[kk], false, bm[kk], (short)0, acc, false, false);
    }
#pragma unroll
    for (int r = 0; r < 8; ++r) {                       // EXEC stays all-ones
        const float dv = acc[r];
        const bool p = dv < bestD[r];
        bestD[r] = p ? dv   : bestD[r];
        bestI[r] = p ? code : bestI[r];
    }
}

// ---------------------------------------------------------------------------
// Main kernel: 256 threads = 8 waves; each wave owns 16 pixels, scans all
// 1024 codes with chained V_WMMA_F32_16X16X4_F32 (fp32-exact distances).
// dist(m, n) = |c_n|^2 - 2 * z_m . c_n   (|z|^2 dropped: argmin-invariant)
// B tiles are double-buffered so each WMMA chain overlaps the next chunk's
// global loads (software pipelining).
// ---------------------------------------------------------------------------
__global__ void __launch_bounds__(256)
vq_main_kernel(const float* __restrict__ x,
               const float* __restrict__ cb,
               const float* __restrict__ c2,
               float* __restrict__ out) {
    __shared__ int s_idx[128];

    const int tid  = threadIdx.x;
    const int lane = tid & 31;
    const int wave = tid >> 5;
    const int m    = lane & 15;   // pixel row (A) / code col (B) within tile
    const int h    = lane >> 4;   // selects K-pair {2h, 2h+1} of each K=4 step

    const int blockPix = blockIdx.x * 128;       // 128 pixels per workgroup
    const int wavePix  = blockPix + wave * 16;   // 16 pixels per wave
    const int b        = wavePix >> 12;          // batch index (HW = 4096)
    const int hw       = (wavePix & (HW - 1)) + m;

    // ---- Preload A tile (16 px x 64 dims), pre-scaled by -2 ----
    // 16x4 f32 A layout: lanes 0-15 hold K=0,1 ; lanes 16-31 hold K=2,3.
    v2f a[16];
    {
        const float* xb = x + (size_t)(b * CODE_DIM) * HW + hw;
#pragma unroll
        for (int kk = 0; kk < 16; ++kk) {
            const int d0 = 4 * kk + 2 * h;
            a[kk][0] = -2.0f * xb[(size_t)d0 * HW];
            a[kk][1] = -2.0f * xb[(size_t)(d0 + 1) * HW];
        }
    }

    float bestD[8];
    int   bestI[8];
#pragma unroll
    for (int r = 0; r < 8; ++r) { bestD[r] = 3.4e38f; bestI[r] = 0; }

    // ---- Scan 64 chunks of 16 codes, double-buffered (unroll by 2) ----
    v2f   b0[16], b1[16];
    float c20, c21;
    load_chunk(cb, c2, 0, m, h, b0, c20);
    for (int c = 0; c < NUM_CODES / 16; c += 2) {
        load_chunk(cb, c2, c + 1, m, h, b1, c21);       // prefetch odd chunk
        process_chunk(c, m, a, b0, c20, bestD, bestI);  // WMMAs hide b1 loads
        if (c + 2 < NUM_CODES / 16)
            load_chunk(cb, c2, c + 2, m, h, b0, c20);   // prefetch even chunk
        process_chunk(c + 1, m, a, b1, c21, bestD, bestI);
    }

    // ---- Argmin across the 16 code-lanes of each half-wave ----
    // D layout: VGPR r -> pixel M = r (lanes 0-15) / M = r+8 (lanes 16-31).
#pragma unroll
    for (int r = 0; r < 8; ++r) {
        float d = bestD[r];
        int   i = bestI[r];
#pragma unroll
        for (int off = 1; off < 16; off <<= 1) {
            const float od = __shfl_xor(d, off, 32);
            const int   oi = __shfl_xor(i, off, 32);
            const bool p = (od < d) || (od == d && oi < i);  // first-min ties
            d = p ? od : d;
            i = p ? oi : i;
        }
        if (m == r) s_idx[wave * 16 + h * 8 + r] = i;
    }
    __syncthreads();

    // ---- Emit indices (as float, per flattened output dtype) ----
    if (tid < 128)
        out[IDX_OFF + blockPix + tid] = (float)s_idx[tid];

    // ---- Gather z_q, store in (B,D,H,W) layout, accumulate loss ----
    const int bb  = blockPix >> 12;
    const int hw0 = blockPix & (HW - 1);
    float lsum = 0.0f;
#pragma unroll
    for (int it = 0; it < 32; ++it) {
        const int i = it * 256 + tid;     // 0..8191 over 128 px x 64 dims
        const int d = i >> 7;             // dim
        const int t = i & 127;            // pixel within block
        const int code = s_idx[t];
        const float zq = cb[code * CODE_DIM + d];
        const size_t off = (size_t)(bb * CODE_DIM + d) * HW + hw0 + t;
        const float xv = x[off];
        out[off] = zq;                    // coalesced 1 KB per iteration
        const float diff = zq - xv;
        lsum += diff * diff;
    }
    // Wave reduction, one atomic per wave; scale folds (1+beta)/N.
#pragma unroll
    for (int off = 16; off >= 1; off >>= 1)
        lsum += __shfl_xor(lsum, off, 32);
    if (lane == 0)
        atomicAdd(out + LOSS_OFF, lsum * (1.25f / 8388608.0f));
}

// ---------------------------------------------------------------------------
extern "C" void kernel_launch(void* const* d_in, const int* in_sizes, int n_in,
                              void* d_out, int out_size, void* d_ws, size_t ws_size,
                              hipStream_t stream) {
    const float* x  = (const float*)d_in[0];   // (32, 64, 64, 64) f32
    const float* cb = (const float*)d_in[1];   // (1024, 64) f32
    float* out = (float*)d_out;
    float* c2  = (float*)d_ws;                 // 1024 floats of scratch

    vq_c2_kernel<<<dim3((NUM_CODES + 255) / 256), dim3(256), 0, stream>>>(cb, c2, out);
    vq_main_kernel<<<dim3(131072 / 128), dim3(256), 0, stream>>>(x, cb, c2, out);
}